// Temporal_Attention_layer_73572789780806
// MI455X (gfx1250) — compile-verified
//
#include <hip/hip_runtime.h>
#include <hip/hip_bf16.h>
#include <math.h>

// Problem constants (match reference)
#define BB 32
#define NN 2048
#define FF 16
#define TT 64
#define SPLIT 16
#define NPER (NN / SPLIT)   // 128 n-values per pass-1 block

typedef __attribute__((ext_vector_type(2))) float v2f;
typedef __attribute__((ext_vector_type(8))) float v8f;

// -----------------------------------------------------------------------------
// Pass 1: single streaming pass over x (256 MB, HBM-bound).
// Computes partial A[b,t,f] = sum_n U1[n] x[b,n,f,t]
//      and partial M[b,f,u] = sum_n U2[f,n] * (sum_f' U3[f'] x[b,n,f',u])
// Each block: one (b, n-chunk). Thread owns (t, n-stride), accumulates in regs.
// Deterministic tree: regs -> LDS reduce over 4 n-lanes -> workspace partial.
// -----------------------------------------------------------------------------
__global__ __launch_bounds__(256) void ta_pass1(
    const float* __restrict__ x, const float* __restrict__ U1,
    const float* __restrict__ U2, const float* __restrict__ U3,
    float* __restrict__ ws)
{
    __shared__ float U2c[FF * NPER];     // 8 KB: U2[f, n0..n0+127]
    __shared__ float U1c[NPER];          // 0.5 KB
    __shared__ float red[4 * TT * FF];   // 16 KB reduction buffer

    const int tid = threadIdx.x;
    const int b  = blockIdx.x / SPLIT;
    const int sp = blockIdx.x % SPLIT;
    const int n0 = sp * NPER;
    const int t  = tid & 63;             // this thread's t (= u) index
    const int nl = tid >> 6;             // n-lane 0..3

    // Stage the U2 / U1 chunks this block needs
    for (int e = tid; e < FF * NPER; e += 256) {
        int f = e >> 7;                  // e / NPER
        int j = e & (NPER - 1);
        U2c[e] = U2[f * NN + n0 + j];
    }
    if (tid < NPER) U1c[tid] = U1[n0 + tid];

    float u3r[FF];
#pragma unroll
    for (int f = 0; f < FF; ++f) u3r[f] = U3[f];   // broadcast -> s_load

    float a_acc[FF], m_acc[FF];
#pragma unroll
    for (int f = 0; f < FF; ++f) { a_acc[f] = 0.0f; m_acc[f] = 0.0f; }

    __syncthreads();

    // Main stream: 32 n-iterations per thread, 16 fp32 loads each.
    for (int j = nl; j < NPER; j += 4) {
        const int n = n0 + j;
        const float u1 = U1c[j];
        const float* xb = x + (size_t)(b * NN + n) * (FF * TT) + t;
        float xv[FF];
#pragma unroll
        for (int f = 0; f < FF; ++f)
            xv[f] = __builtin_nontemporal_load(&xb[f * TT]);  // streaming: don't pollute caches

        float tmp = 0.0f;
#pragma unroll
        for (int f = 0; f < FF; ++f) tmp += u3r[f] * xv[f];   // rhs[b,n,t]
#pragma unroll
        for (int f = 0; f < FF; ++f) a_acc[f] += u1 * xv[f];  // A[t,f] partial
#pragma unroll
        for (int f = 0; f < FF; ++f) m_acc[f] += U2c[f * NPER + j] * tmp;  // M[f,t] partial
    }

    float* outp = ws + (size_t)(b * SPLIT + sp) * 2048;

    // Reduce A over the 4 n-lanes, store [t][f]
#pragma unroll
    for (int f = 0; f < FF; ++f) red[(nl * TT + t) * FF + f] = a_acc[f];
    __syncthreads();
    for (int e = tid; e < TT * FF; e += 256) {
        float v = red[e] + red[1024 + e] + red[2048 + e] + red[3072 + e];
        outp[e] = v;                                   // A partial, [t][f]
    }
    __syncthreads();

    // Reduce M over the 4 n-lanes, store [f][u]  (reg layout was [u][f])
#pragma unroll
    for (int f = 0; f < FF; ++f) red[(nl * TT + t) * FF + f] = m_acc[f];
    __syncthreads();
    for (int e = tid; e < TT * FF; e += 256) {
        int f = e >> 6;
        int u = e & 63;
        int idx = u * FF + f;
        float v = red[idx] + red[1024 + idx] + red[2048 + idx] + red[3072 + idx];
        outp[1024 + e] = v;                            // M partial, [f][u]
    }
}

// -----------------------------------------------------------------------------
// Pass 2: one block per batch b (8 waves).
//   1) reduce SPLIT partials -> A (64x16), M (16x64) in LDS
//   2) product = A @ M  via v_wmma_f32_16x16x4_f32 ; sigmoid(product + be)
//   3) E = Ve @ S       via v_wmma_f32_16x16x4_f32
//   4) softmax over axis=1 (rows t) per column u
// WMMA fp32 lane layout (ISA 7.12.2):
//   A 16x4 : lane m (0..15 | 16..31), vgpr0 = K {0|2}, vgpr1 = K {1|3}
//   B 4x16 : lanes 0-15 hold K {0,1}, lanes 16-31 hold K {2,3}, col = lane%16
//   C 16x16: vgpr r, lanes 0-15 -> row r, lanes 16-31 -> row r+8, col = lane%16
// -----------------------------------------------------------------------------
__global__ __launch_bounds__(256) void ta_pass2(
    const float* __restrict__ ws, const float* __restrict__ be,
    const float* __restrict__ Ve, float* __restrict__ out)
{
    __shared__ float Al[TT * 17];   // A [t][f], padded stride 17 (bank-friendly)
    __shared__ float Ml[FF * 65];   // M [f][u], padded stride 65
    __shared__ float Vl[TT * 65];   // Ve [t][s]
    __shared__ float Sm[TT * 65];   // sigmoid(product+be) [s][u]
    __shared__ float El[TT * 65];   // E [t][u]

    const int tid = threadIdx.x;
    const int b  = blockIdx.x;
    const int w  = tid >> 5;        // wave id 0..7
    const int l  = tid & 31;
    const int m  = l & 15;          // lane-in-half
    const int hi = l >> 4;          // 0: lanes 0-15, 1: lanes 16-31

    // 1) reduce partials (fixed order -> deterministic)
    const float* wsb = ws + (size_t)b * SPLIT * 2048;
    for (int e = tid; e < 2048; e += 256) {
        float acc = 0.0f;
#pragma unroll
        for (int sp = 0; sp < SPLIT; ++sp) acc += wsb[sp * 2048 + e];
        if (e < 1024) {
            Al[(e >> 4) * 17 + (e & 15)] = acc;
        } else {
            int e2 = e - 1024;
            Ml[(e2 >> 6) * 65 + (e2 & 63)] = acc;
        }
    }
    for (int e = tid; e < TT * TT; e += 256)
        Vl[(e >> 6) * 65 + (e & 63)] = Ve[e];
    __syncthreads();

    // 2) product = A(64x16) @ M(16x64); 16 tiles, 2 per wave; K=16 -> 4 wmma steps
#pragma unroll
    for (int tt = 0; tt < 2; ++tt) {
        const int tile = w + tt * 8;
        const int ti = tile >> 2, tj = tile & 3;
        v8f acc = {};
#pragma unroll
        for (int kk = 0; kk < 4; ++kk) {
            const int k0 = 4 * kk + 2 * hi;
            v2f a, bm;
            a.x  = Al[(ti * 16 + m) * 17 + k0];
            a.y  = Al[(ti * 16 + m) * 17 + k0 + 1];
            bm.x = Ml[k0 * 65 + tj * 16 + m];
            bm.y = Ml[(k0 + 1) * 65 + tj * 16 + m];
            acc = __builtin_amdgcn_wmma_f32_16x16x4_f32(
                false, a, false, bm, (short)0, acc, false, false);
        }
#pragma unroll
        for (int r = 0; r < 8; ++r) {
            const int row = ti * 16 + r + 8 * hi;
            const int col = tj * 16 + m;
            const float v = acc[r] + be[row * 64 + col];
            Sm[row * 65 + col] = 1.0f / (1.0f + __expf(-v));
        }
    }
    __syncthreads();

    // 3) E = Ve(64x64) @ S(64x64); K=64 -> 16 wmma steps per tile
#pragma unroll
    for (int tt = 0; tt < 2; ++tt) {
        const int tile = w + tt * 8;
        const int ti = tile >> 2, tj = tile & 3;
        v8f acc = {};
        for (int kk = 0; kk < 16; ++kk) {
            const int k0 = 4 * kk + 2 * hi;
            v2f a, bm;
            a.x  = Vl[(ti * 16 + m) * 65 + k0];
            a.y  = Vl[(ti * 16 + m) * 65 + k0 + 1];
            bm.x = Sm[k0 * 65 + tj * 16 + m];
            bm.y = Sm[(k0 + 1) * 65 + tj * 16 + m];
            acc = __builtin_amdgcn_wmma_f32_16x16x4_f32(
                false, a, false, bm, (short)0, acc, false, false);
        }
#pragma unroll
        for (int r = 0; r < 8; ++r) {
            const int row = ti * 16 + r + 8 * hi;
            const int col = tj * 16 + m;
            El[row * 65 + col] = acc[r];
        }
    }
    __syncthreads();

    // 4) softmax over t (axis=1) per column u; each of 64 threads owns a column
    if (tid < TT) {
        const int u = tid;
        float mx = -INFINITY;
        for (int t = 0; t < TT; ++t) mx = fmaxf(mx, El[t * 65 + u]);
        float sum = 0.0f;
        for (int t = 0; t < TT; ++t) {
            float e = __expf(El[t * 65 + u] - mx);
            El[t * 65 + u] = e;       // thread owns this column exclusively
            sum += e;
        }
        const float inv = 1.0f / sum;
        for (int t = 0; t < TT; ++t)
            out[((size_t)b * TT + t) * TT + u] = El[t * 65 + u] * inv;
    }
}

extern "C" void kernel_launch(void* const* d_in, const int* in_sizes, int n_in,
                              void* d_out, int out_size, void* d_ws, size_t ws_size,
                              hipStream_t stream) {
    const float* x  = (const float*)d_in[0];
    const float* U1 = (const float*)d_in[1];
    const float* U2 = (const float*)d_in[2];
    const float* U3 = (const float*)d_in[3];
    const float* be = (const float*)d_in[4];
    const float* Ve = (const float*)d_in[5];
    float* out = (float*)d_out;
    float* ws  = (float*)d_ws;   // needs BB*SPLIT*2048*4 = 4 MB

    ta_pass1<<<dim3(BB * SPLIT), dim3(256), 0, stream>>>(x, U1, U2, U3, ws);
    ta_pass2<<<dim3(BB), dim3(256), 0, stream>>>(ws, be, Ve, out);
}